// HighLevelGuidedFrequencyFusion_12816182411645
// MI455X (gfx1250) — compile-verified
//
#include <hip/hip_runtime.h>
#include <math.h>

// ---------------------------------------------------------------------------
// HighLevelGuidedFrequencyFusion for MI455X (gfx1250), all-f32 with
// V_WMMA_F32_16X16X4_F32 for every GEMM.  Memory-bound workload (~2-3 GB
// traffic vs ~45 GFLOP) -> keep f32 precision, WMMA f32 path.
//
// Input order assumption: setup_inputs() dict order {vis_feat, ir_feat,
// params}, with the nested params pytree flattened with sorted keys at every
// level (uppercase sorts before lowercase).
// ---------------------------------------------------------------------------

#define B_     2
#define C_     64
#define H_     256
#define W_     256
#define NTOK   4096      // (H/4)*(W/4)
#define KSEL   1024      // top-k = NTOK/4
#define TOK    1024      // C*4*4
#define EMB_   128
#define HEADS_ 4
#define HD_    32
#define PLANE_ ((size_t)B_*C_*H_*W_)   // 8388608

typedef __attribute__((ext_vector_type(2))) float v2f;
typedef __attribute__((ext_vector_type(8))) float v8f;

__device__ __forceinline__ float geluf(float x) {
  float x3 = x * x * x;
  return 0.5f * x * (1.f + tanhf(0.7978845608028654f * (x + 0.044715f * x3)));
}

// ------------------------------- reductions --------------------------------
__global__ void mean_kernel(const float* __restrict__ vis,
                            const float* __restrict__ ir,
                            float* __restrict__ g) {
  int blk = blockIdx.x;            // B*128 blocks
  int b = blk >> 7, c = blk & 127;
  const float* src = (c < C_) ? (vis + ((size_t)(b * C_ + c) << 16))
                              : (ir  + ((size_t)(b * C_ + (c - C_)) << 16));
  __shared__ float red[256];
  float s = 0.f;
  for (int i = threadIdx.x; i < H_ * W_; i += 256) s += src[i];
  red[threadIdx.x] = s; __syncthreads();
  for (int o = 128; o > 0; o >>= 1) {
    if (threadIdx.x < o) red[threadIdx.x] += red[threadIdx.x + o];
    __syncthreads();
  }
  if (threadIdx.x == 0) g[b * 128 + c] = red[0] * (1.f / (H_ * W_));
}

// -------------------------- intent MLP chain (tiny) ------------------------
__global__ void intent_kernel(
    const float* g, const float* Wr, const float* br, const float* prompts,
    const float* Wf1, const float* bf1, const float* lng, const float* lnb,
    const float* Wf2, const float* bf2,
    const float* Wa1, const float* ba1, const float* Wa2, const float* ba2,
    const float* WiScA, const float* WintA, const float* WiBypA,
    const float* WiScP, const float* WintP, const float* WiBypP,
    float* fi_out, float* gamma, float* beta,
    float* iScA, float* iIntA, float* iBypA,
    float* iScP, float* iIntP, float* iBypP) {
  __shared__ float gv[128], qv[64], lv[8], wv[8], tiv[64], tv[64], hv[64],
      fiv[64], av[128], stat[2];
  int e = threadIdx.x;
  for (int b = 0; b < B_; b++) {
    gv[e] = g[b * 128 + e];
    __syncthreads();
    if (e < 64) { float s = br[e]; for (int i = 0; i < 128; i++) s += gv[i] * Wr[i * 64 + e]; qv[e] = geluf(s); }
    __syncthreads();
    if (e < 5) { float s = 0; for (int i = 0; i < 64; i++) s += qv[i] * prompts[e * 64 + i]; lv[e] = s * 0.125f; }
    __syncthreads();
    if (e == 0) {
      float mx = -1e30f; for (int p = 0; p < 5; p++) mx = fmaxf(mx, lv[p]);
      float ss = 0; for (int p = 0; p < 5; p++) { wv[p] = expf(lv[p] - mx); ss += wv[p]; }
      for (int p = 0; p < 5; p++) wv[p] /= ss;
    }
    __syncthreads();
    if (e < 64) { float s = 0; for (int p = 0; p < 5; p++) s += wv[p] * prompts[p * 64 + e]; tiv[e] = s; }
    __syncthreads();
    if (e < 64) { float s = bf1[e]; for (int i = 0; i < 64; i++) s += tiv[i] * Wf1[i * 64 + e]; tv[e] = s; }
    __syncthreads();
    if (e == 0) {
      float m = 0; for (int i = 0; i < 64; i++) m += tv[i]; m *= (1.f / 64.f);
      float v = 0; for (int i = 0; i < 64; i++) { float d = tv[i] - m; v += d * d; }
      stat[0] = m; stat[1] = rsqrtf(v * (1.f / 64.f) + 1e-5f);
    }
    __syncthreads();
    if (e < 64) hv[e] = geluf((tv[e] - stat[0]) * stat[1] * lng[e] + lnb[e]);
    __syncthreads();
    if (e < 64) { float s = bf2[e]; for (int i = 0; i < 64; i++) s += hv[i] * Wf2[i * 64 + e]; fiv[e] = s; fi_out[b * 64 + e] = s; }
    __syncthreads();
    { float s = ba1[e]; for (int i = 0; i < 64; i++) s += fiv[i] * Wa1[i * 128 + e]; av[e] = geluf(s); }
    __syncthreads();
    { float s = ba2[e]; for (int i = 0; i < 128; i++) s += av[i] * Wa2[i * 128 + e];
      if (e < 64) gamma[b * 64 + e] = s; else beta[b * 64 + e - 64] = s; }
    { float s;
      s = 0; for (int i = 0; i < 64; i++) s += fiv[i] * WiScA [i * 128 + e]; iScA [b * 128 + e] = s;
      s = 0; for (int i = 0; i < 64; i++) s += fiv[i] * WintA [i * 128 + e]; iIntA[b * 128 + e] = s;
      s = 0; for (int i = 0; i < 64; i++) s += fiv[i] * WiBypA[i * 128 + e]; iBypA[b * 128 + e] = s;
      s = 0; for (int i = 0; i < 64; i++) s += fiv[i] * WiScP [i * 128 + e]; iScP [b * 128 + e] = s;
      s = 0; for (int i = 0; i < 64; i++) s += fiv[i] * WintP [i * 128 + e]; iIntP[b * 128 + e] = s;
      s = 0; for (int i = 0; i < 64; i++) s += fiv[i] * WiBypP[i * 128 + e]; iBypP[b * 128 + e] = s; }
    __syncthreads();
  }
}

// ------------------------------ 256-point FFT ------------------------------
__device__ __forceinline__ void fft256_lds(float2* s, int tid, float sign) {
  for (int i = tid; i < 256; i += 128) {
    int r = (int)(__brev((unsigned)i) >> 24);
    if (r > i) { float2 t = s[i]; s[i] = s[r]; s[r] = t; }
  }
  __syncthreads();
  for (int len = 2; len <= 256; len <<= 1) {
    int m = len >> 1;
    int jj = tid % m;
    int base = (tid / m) * len;
    float ang = sign * 6.283185307179586f * (float)jj / (float)len;
    float cw = cosf(ang), sw = sinf(ang);
    float2 a = s[base + jj];
    float2 b = s[base + jj + m];
    float2 t; t.x = b.x * cw - b.y * sw; t.y = b.x * sw + b.y * cw;
    s[base + jj]     = make_float2(a.x + t.x, a.y + t.y);
    s[base + jj + m] = make_float2(a.x - t.x, a.y - t.y);
    __syncthreads();
  }
}

// Row-FFT + transposed write (pass is used for both fwd pass1 and inv pass1).
__global__ __launch_bounds__(128) void fft_pass(const float* __restrict__ in_re,
                                                const float* __restrict__ in_im,
                                                float* __restrict__ out_re,
                                                float* __restrict__ out_im,
                                                float sign) {
  __shared__ float2 s[256];
  int row = blockIdx.x, p = blockIdx.y, tid = threadIdx.x;
  size_t base = ((size_t)p << 16) + ((size_t)row << 8);
  for (int i = tid; i < 256; i += 128) {
    float2 v; v.x = in_re[base + i]; v.y = in_im ? in_im[base + i] : 0.f; s[i] = v;
  }
  __syncthreads();
  fft256_lds(s, tid, sign);
  for (int i = tid; i < 256; i += 128) {
    size_t o = ((size_t)p << 16) + ((size_t)i << 8) + row;
    out_re[o] = s[i].x; out_im[o] = s[i].y;
  }
}

// Second forward pass: FFT + amplitude/phase, transposed write -> natural.
__global__ __launch_bounds__(128) void fft_ampph(const float* __restrict__ in_re,
                                                 const float* __restrict__ in_im,
                                                 float* __restrict__ amp,
                                                 float* __restrict__ ph) {
  __shared__ float2 s[256];
  int row = blockIdx.x, p = blockIdx.y, tid = threadIdx.x;
  size_t base = ((size_t)p << 16) + ((size_t)row << 8);
  for (int i = tid; i < 256; i += 128) { float2 v; v.x = in_re[base + i]; v.y = in_im[base + i]; s[i] = v; }
  __syncthreads();
  fft256_lds(s, tid, -1.f);
  for (int i = tid; i < 256; i += 128) {
    size_t o = ((size_t)p << 16) + ((size_t)i << 8) + row;
    amp[o] = sqrtf(s[i].x * s[i].x + s[i].y * s[i].y);
    ph[o]  = atan2f(s[i].y, s[i].x);
  }
}

// Second inverse pass: FFT(+1) + real part with 1/(H*W) scale.
__global__ __launch_bounds__(128) void fft_real(const float* __restrict__ in_re,
                                                const float* __restrict__ in_im,
                                                float* __restrict__ out) {
  __shared__ float2 s[256];
  int row = blockIdx.x, p = blockIdx.y, tid = threadIdx.x;
  size_t base = ((size_t)p << 16) + ((size_t)row << 8);
  for (int i = tid; i < 256; i += 128) { float2 v; v.x = in_re[base + i]; v.y = in_im[base + i]; s[i] = v; }
  __syncthreads();
  fft256_lds(s, tid, 1.f);
  for (int i = tid; i < 256; i += 128) {
    size_t o = ((size_t)p << 16) + ((size_t)i << 8) + row;
    out[o] = s[i].x * (1.f / 65536.f);
  }
}

// ------------------------- patchify / unpatchify ---------------------------
__device__ __forceinline__ size_t patch_plane_idx(size_t gid) {
  int f = gid & 1023;
  int n = (int)((gid >> 10) & 4095);
  int b = (int)(gid >> 22);
  int c = f >> 4, py = (f >> 2) & 3, px = f & 3;
  int gy = n >> 6, gx = n & 63;
  return (((size_t)(b * C_ + c)) << 16) + (size_t)(gy * 4 + py) * 256 + (gx * 4 + px);
}

__global__ void patchify_k(const float* __restrict__ plane, float* __restrict__ tok) {
  size_t gid = (size_t)blockIdx.x * blockDim.x + threadIdx.x;
  if (gid >= (size_t)B_ * NTOK * TOK) return;
  tok[gid] = plane[patch_plane_idx(gid)];
}

__global__ void unpatchify_k(const float* __restrict__ tok, float* __restrict__ plane) {
  size_t gid = (size_t)blockIdx.x * blockDim.x + threadIdx.x;
  if (gid >= (size_t)B_ * NTOK * TOK) return;
  plane[patch_plane_idx(gid)] = tok[gid];
}

__global__ void coordw_k(const float* __restrict__ Wc, float* __restrict__ cw) {
  int gid = blockIdx.x * blockDim.x + threadIdx.x;
  if (gid >= NTOK * 128) return;
  int e = gid & 127, n = gid >> 7;
  float cy = (float)(n >> 6) * (1.f / 63.f);
  float cx = (float)(n & 63) * (1.f / 63.f);
  cw[gid] = cy * Wc[e] + cx * Wc[128 + e];
}

// ------------------------------ WMMA f32 GEMM ------------------------------
// C[M,N] = gelu?( alpha*A·B + bias[n] + bvec[row/rpb][n] + cw[row%rpb][n]
//                 + e0s*E0 + e1s*E1 )
// amode: 0 plain A; 1 A=[A|A2] (Ksub each); 2 A=[A|A2||A-A2|] (Ksub each)
// btrans: B element (k,n) taken from Bm[n*ldb + k]
__global__ __launch_bounds__(256) void gemm_f32_wmma(
    const float* __restrict__ A, const float* __restrict__ A2, int amode,
    int Ksub, int lda, const float* __restrict__ Bm, int ldb, int btrans,
    float* __restrict__ Cc, int ldc, int M, int N, int Kdim, float alpha,
    const float* __restrict__ bias, const float* __restrict__ bvec, int rpb,
    const float* __restrict__ cw, int ldaux,
    const float* __restrict__ e0, float e0s,
    const float* __restrict__ e1, float e1s, int ldE, int do_gelu) {
  __shared__ float As[32][33];
  __shared__ float Bs[32][65];
  int tid = threadIdx.x;
  int bm = blockIdx.y * 32;
  int bn = blockIdx.x * 64;
  int wave = tid >> 5, lane = tid & 31;
  int wm = wave >> 2, wn = wave & 3;
  v8f acc = {0.f, 0.f, 0.f, 0.f, 0.f, 0.f, 0.f, 0.f};

  for (int k0 = 0; k0 < Kdim; k0 += 32) {
    for (int i = tid; i < 32 * 32; i += 256) {
      int r = i >> 5, c = i & 31;
      int grow = bm + r, kk = k0 + c;
      float v = 0.f;
      if (grow < M && kk < Kdim) {
        if (amode == 0) v = A[(size_t)grow * lda + kk];
        else if (amode == 1)
          v = (kk < Ksub) ? A[(size_t)grow * lda + kk]
                          : A2[(size_t)grow * lda + kk - Ksub];
        else {
          if (kk < Ksub) v = A[(size_t)grow * lda + kk];
          else if (kk < 2 * Ksub) v = A2[(size_t)grow * lda + kk - Ksub];
          else v = fabsf(A[(size_t)grow * lda + kk - 2 * Ksub] -
                         A2[(size_t)grow * lda + kk - 2 * Ksub]);
        }
      }
      As[r][c] = v;
    }
    for (int i = tid; i < 32 * 64; i += 256) {
      int r = i >> 6, c = i & 63;
      int kk = k0 + r, gn = bn + c;
      float v = 0.f;
      if (kk < Kdim && gn < N)
        v = btrans ? Bm[(size_t)gn * ldb + kk] : Bm[(size_t)kk * ldb + gn];
      Bs[r][c] = v;
    }
    __syncthreads();
    // A 16x4 frag: lanes0-15 M=lane K={kb,kb+1}; lanes16-31 K={kb+2,kb+3}
    int mrow = wm * 16 + (lane & 15);
    int ncol = wn * 16 + (lane & 15);
    int kh = (lane >> 4) * 2;
#pragma unroll
    for (int kb = 0; kb < 32; kb += 4) {
      v2f a, b;
      a.x = As[mrow][kb + kh];     a.y = As[mrow][kb + kh + 1];
      b.x = Bs[kb + kh][ncol];     b.y = Bs[kb + kh + 1][ncol];
      acc = __builtin_amdgcn_wmma_f32_16x16x4_f32(false, a, false, b,
                                                  (short)0, acc, false, false);
    }
    __syncthreads();
  }
  // C layout: VGPR r -> M = r + 8*(lane>=16), N = lane&15
  int rbase = bm + wm * 16 + ((lane >> 4) * 8);
  int gcol = bn + wn * 16 + (lane & 15);
  if (gcol < N) {
#pragma unroll
    for (int r = 0; r < 8; r++) {
      int grow = rbase + r;
      if (grow < M) {
        float v = acc[r] * alpha;
        if (bias) v += bias[gcol];
        if (bvec) v += bvec[(grow / rpb) * ldaux + gcol];
        if (cw)   v += cw[(size_t)(grow % rpb) * ldaux + gcol];
        if (e0)   v += e0s * e0[(size_t)grow * ldE + gcol];
        if (e1)   v += e1s * e1[(size_t)grow * ldE + gcol];
        if (do_gelu) v = geluf(v);
        Cc[(size_t)grow * ldc + gcol] = v;
      }
    }
  }
}

// ------------------------------- score head --------------------------------
__global__ void score_reduce_k(const float* __restrict__ h,
                               const float* __restrict__ Wo,
                               const float* __restrict__ bo,
                               float* __restrict__ score, int rows) {
  int row = blockIdx.x * blockDim.x + threadIdx.x;
  if (row >= rows) return;
  float s = bo[0];
  for (int e = 0; e < 128; e++) s += h[(size_t)row * 128 + e] * Wo[e];
  score[row] = s;
}

// --------------------------- top-k (bitonic sort) --------------------------
// forward value of routing == hard mask; only the SET of top-k indices
// matters (gather/scatter are index-consistent), so any exact-k select works.
__global__ __launch_bounds__(1024) void topk_k(const float* __restrict__ score,
                                               int* __restrict__ topi) {
  __shared__ float vals[4096];
  __shared__ int idxs[4096];
  int b = blockIdx.x, tid = threadIdx.x;
  for (int i = tid; i < 4096; i += 1024) { vals[i] = score[b * 4096 + i]; idxs[i] = i; }
  __syncthreads();
  for (int k = 2; k <= 4096; k <<= 1)
    for (int j = k >> 1; j > 0; j >>= 1) {
      for (int i = tid; i < 4096; i += 1024) {
        int l = i ^ j;
        if (l > i) {
          bool sw = ((i & k) == 0) ? (vals[i] < vals[l]) : (vals[i] > vals[l]);
          if (sw) {
            float tv = vals[i]; vals[i] = vals[l]; vals[l] = tv;
            int ti = idxs[i]; idxs[i] = idxs[l]; idxs[l] = ti;
          }
        }
      }
      __syncthreads();
    }
  if (tid < 1024) topi[b * 1024 + tid] = idxs[tid];  // descending: first k
}

__global__ void gather_k(const float* __restrict__ visT,
                         const float* __restrict__ irT,
                         const int* __restrict__ topi,
                         float* __restrict__ selV, float* __restrict__ selI) {
  size_t gid = (size_t)blockIdx.x * blockDim.x + threadIdx.x;
  if (gid >= (size_t)B_ * KSEL * TOK) return;
  int f = gid & 1023, ks = (int)((gid >> 10) & 1023), b = (int)(gid >> 20);
  int n = topi[b * 1024 + ks];
  size_t src = ((size_t)(b * NTOK + n)) * 1024 + f;
  selV[gid] = visT[src]; selI[gid] = irT[src];
}

__global__ void scatter_k(const float* __restrict__ strong,
                          const int* __restrict__ topi,
                          float* __restrict__ fusedT) {
  size_t gid = (size_t)blockIdx.x * blockDim.x + threadIdx.x;
  if (gid >= (size_t)B_ * KSEL * TOK) return;
  int f = gid & 1023, ks = (int)((gid >> 10) & 1023), b = (int)(gid >> 20);
  int n = topi[b * 1024 + ks];
  fusedT[((size_t)(b * NTOK + n)) * 1024 + f] = strong[gid];
}

// ------------------------------- softmax row -------------------------------
__global__ void softmax_k(float* __restrict__ att) {
  __shared__ float red[256];
  float* p = att + (size_t)blockIdx.x * 1024;
  float mx = -1e30f;
  for (int i = threadIdx.x; i < 1024; i += 256) mx = fmaxf(mx, p[i]);
  red[threadIdx.x] = mx; __syncthreads();
  for (int o = 128; o > 0; o >>= 1) {
    if (threadIdx.x < o) red[threadIdx.x] = fmaxf(red[threadIdx.x], red[threadIdx.x + o]);
    __syncthreads();
  }
  mx = red[0]; __syncthreads();
  float s = 0.f;
  for (int i = threadIdx.x; i < 1024; i += 256) { float e = expf(p[i] - mx); p[i] = e; s += e; }
  red[threadIdx.x] = s; __syncthreads();
  for (int o = 128; o > 0; o >>= 1) {
    if (threadIdx.x < o) red[threadIdx.x] += red[threadIdx.x + o];
    __syncthreads();
  }
  float inv = 1.f / red[0];
  for (int i = threadIdx.x; i < 1024; i += 256) p[i] *= inv;
}

// ------------------------------- spec & final ------------------------------
// NOTE: arctan2(sin x, cos x) followed by exp(i*.) is an identity -> skipped.
__global__ void spec_k(const float* __restrict__ amp, const float* __restrict__ ph,
                       float* __restrict__ sre, float* __restrict__ sim) {
  size_t gid = (size_t)blockIdx.x * blockDim.x + threadIdx.x;
  if (gid >= PLANE_) return;
  float a = amp[gid], p = ph[gid];
  sre[gid] = a * cosf(p); sim[gid] = a * sinf(p);
}

__global__ void final_k(const float* __restrict__ r2, const float* __restrict__ gamma,
                        const float* __restrict__ beta, const float* __restrict__ vis,
                        const float* __restrict__ ir, float* __restrict__ out) {
  size_t gid = (size_t)blockIdx.x * blockDim.x + threadIdx.x;
  if (gid >= PLANE_) return;
  int c = (int)((gid >> 16) & 63), b = (int)(gid >> 22);
  float gm = gamma[b * 64 + c], bt = beta[b * 64 + c];
  out[gid] = r2[gid] * (1.f + 0.1f * tanhf(gm)) + 0.1f * bt + 0.5f * (vis[gid] + ir[gid]);
}

// --------------------------------- conv3x3 ---------------------------------
__global__ __launch_bounds__(256) void conv3x3_k(const float* __restrict__ in,
                                                 const float* __restrict__ w,
                                                 const float* __restrict__ bias,
                                                 float* __restrict__ out, int relu) {
  __shared__ float tile[18][18];
  __shared__ float wl[64 * 9];
  int tx = threadIdx.x, ty = threadIdx.y;
  int tid = ty * 16 + tx;
  int bx0 = blockIdx.x * 16, by0 = blockIdx.y * 16, b = blockIdx.z;
  float acc[64];
#pragma unroll
  for (int i = 0; i < 64; i++) acc[i] = 0.f;
  for (int ci = 0; ci < 64; ci++) {
    const float* ip = in + ((size_t)(b * 64 + ci) << 16);
    for (int i = tid; i < 18 * 18; i += 256) {
      int r = i / 18, c = i % 18;
      int y = by0 + r - 1, x = bx0 + c - 1;
      tile[r][c] = (y >= 0 && y < 256 && x >= 0 && x < 256) ? ip[y * 256 + x] : 0.f;
    }
    for (int i = tid; i < 576; i += 256) wl[i] = w[(i / 9) * 576 + ci * 9 + (i % 9)];
    __syncthreads();
    float win[9];
#pragma unroll
    for (int dy = 0; dy < 3; dy++)
#pragma unroll
      for (int dx = 0; dx < 3; dx++) win[dy * 3 + dx] = tile[ty + dy][tx + dx];
#pragma unroll
    for (int co = 0; co < 64; co++) {
      float s = 0.f;
#pragma unroll
      for (int q = 0; q < 9; q++) s += win[q] * wl[co * 9 + q];
      acc[co] += s;
    }
    __syncthreads();
  }
#pragma unroll
  for (int co = 0; co < 64; co++) {
    float v = acc[co] + bias[co];
    if (relu) v = fmaxf(v, 0.f);
    out[((size_t)(b * 64 + co) << 16) + (by0 + ty) * 256 + (bx0 + tx)] = v;
  }
}

// ------------------------------- host side ---------------------------------
enum {
  IN_VIS = 0, IN_IR, IN_Wa1, IN_Wa2, IN_Wc1, IN_Wc2, IN_Wf1, IN_Wf2, IN_Wr,
  IN_AB_W1, IN_AB_W2, IN_AB_Wi, IN_AB_b1,
  IN_AI_Wb, IN_AI_We_i, IN_AI_We_v, IN_AI_Wint, IN_AI_Wk, IN_AI_Wo, IN_AI_Wq, IN_AI_Wv,
  IN_AS_Wc, IN_AS_Wi, IN_AS_Wo, IN_AS_Wt, IN_AS_bo, IN_AS_bt,
  IN_ba1, IN_ba2, IN_bc1, IN_bc2, IN_bf1, IN_bf2, IN_br, IN_lnb, IN_lng,
  IN_PB_W1, IN_PB_W2, IN_PB_Wi, IN_PB_b1,
  IN_PI_Wb, IN_PI_We_i, IN_PI_We_v, IN_PI_Wint, IN_PI_Wk, IN_PI_Wo, IN_PI_Wq, IN_PI_Wv,
  IN_PS_Wc, IN_PS_Wi, IN_PS_Wo, IN_PS_Wt, IN_PS_bo, IN_PS_bt,
  IN_prompts
};

static inline void launch_gemm(hipStream_t st, const float* A, const float* A2,
                               int amode, int Ksub, int lda, const float* Bm,
                               int ldb, int btrans, float* C, int ldc, int M,
                               int N, int K, float alpha = 1.f,
                               const float* bias = nullptr,
                               const float* bvec = nullptr, int rpb = 1,
                               const float* cw = nullptr, int ldaux = 128,
                               const float* e0 = nullptr, float e0s = 0.f,
                               const float* e1 = nullptr, float e1s = 0.f,
                               int ldE = 0, int gelu = 0) {
  dim3 g((N + 63) / 64, (M + 31) / 32);
  gemm_f32_wmma<<<g, 256, 0, st>>>(A, A2, amode, Ksub, lda, Bm, ldb, btrans, C,
                                   ldc, M, N, K, alpha, bias, bvec, rpb, cw,
                                   ldaux, e0, e0s, e1, e1s, ldE, gelu);
}

struct BranchW {
  const float *Wt, *bt, *Wc, *Wo_s, *bo;                 // score
  const float *We_v, *We_i, *Wq, *Wk, *Wv, *Wo_i, *Wb;   // interact
  const float *W1, *b1, *W2;                             // bypass
  const float *iSc, *iInt, *iByp;                        // intent projections
};

extern "C" void kernel_launch(void* const* d_in, const int* in_sizes, int n_in,
                              void* d_out, int out_size, void* d_ws,
                              size_t ws_size, hipStream_t stream) {
  const float* vis = (const float*)d_in[IN_VIS];
  const float* ir  = (const float*)d_in[IN_IR];
  auto F = [&](int i) { return (const float*)d_in[i]; };

  // ---- workspace arena (floats); ~341 MB total ----
  float* W = (float*)d_ws;
  const size_t P = PLANE_;
  float* amp_v  = W + 0 * P;           // later: fused amplitude plane
  float* ph_v   = W + 1 * P;           // later: fused phase plane
  float* amp_i  = W + 2 * P;           // later: ifft pass-1 scratch (re)
  float* ph_i   = W + 3 * P;           // later: ifft pass-1 scratch (im)
  float* t0     = W + 4 * P;           // fft scratch / attention / spec re
  float* t1     = W + 5 * P;           // fft scratch / spec im
  float* visT   = W + 6 * P;           // later: r0 (ifft real)
  float* irT    = W + 7 * P;           // later: r1 (conv1 out)
  float* fusedT = W + 8 * P;           // later: r2 (conv2 out)
  float* m      = W + 9 * P;
  float* g      = m;            m += 256;
  float* fi     = m;            m += 128;
  float* gamma  = m;            m += 128;
  float* beta   = m;            m += 128;
  float* iScA   = m;            m += 256;
  float* iIntA  = m;            m += 256;
  float* iBypA  = m;            m += 256;
  float* iScP   = m;            m += 256;
  float* iIntP  = m;            m += 256;
  float* iBypP  = m;            m += 256;
  float* cwbuf  = m;            m += (size_t)NTOK * 128;
  float* score  = m;            m += (size_t)B_ * NTOK;
  int*   topi   = (int*)m;      m += (size_t)B_ * KSEL;
  float* hbuf   = m;            m += (size_t)B_ * NTOK * 128;
  float* qin    = m;            m += (size_t)B_ * KSEL * 128;
  float* kin    = m;            m += (size_t)B_ * KSEL * 128;
  float* qm     = m;            m += (size_t)B_ * KSEL * 128;
  float* km     = m;            m += (size_t)B_ * KSEL * 128;
  float* vm     = m;            m += (size_t)B_ * KSEL * 128;
  float* obuf   = m;            m += (size_t)B_ * KSEL * 128;
  float* o2     = m;            m += (size_t)B_ * KSEL * 128;
  float* selV   = m;            m += (size_t)B_ * KSEL * TOK;
  float* selI   = m;            m += (size_t)B_ * KSEL * TOK;
  float* strong = m;            m += (size_t)B_ * KSEL * TOK;
  float* att    = t0;                  // B*4*1024*1024 == PLANE_, aliases t0

  // ---- intent chain ----
  mean_kernel<<<B_ * 128, 256, 0, stream>>>(vis, ir, g);
  intent_kernel<<<1, 128, 0, stream>>>(
      g, F(IN_Wr), F(IN_br), F(IN_prompts), F(IN_Wf1), F(IN_bf1), F(IN_lng),
      F(IN_lnb), F(IN_Wf2), F(IN_bf2), F(IN_Wa1), F(IN_ba1), F(IN_Wa2),
      F(IN_ba2), F(IN_AS_Wi), F(IN_AI_Wint), F(IN_AB_Wi), F(IN_PS_Wi),
      F(IN_PI_Wint), F(IN_PB_Wi), fi, gamma, beta, iScA, iIntA, iBypA, iScP,
      iIntP, iBypP);

  // ---- forward FFT2 -> amplitude & phase planes ----
  dim3 fgrid(H_, B_ * C_);
  fft_pass<<<fgrid, 128, 0, stream>>>(vis, nullptr, t0, t1, -1.f);
  fft_ampph<<<fgrid, 128, 0, stream>>>(t0, t1, amp_v, ph_v);
  fft_pass<<<fgrid, 128, 0, stream>>>(ir, nullptr, t0, t1, -1.f);
  fft_ampph<<<fgrid, 128, 0, stream>>>(t0, t1, amp_i, ph_i);

  // ---- fuse branches (amp, phase) ----
  BranchW bw[2];
  bw[0] = {F(IN_AS_Wt), F(IN_AS_bt), F(IN_AS_Wc), F(IN_AS_Wo), F(IN_AS_bo),
           F(IN_AI_We_v), F(IN_AI_We_i), F(IN_AI_Wq), F(IN_AI_Wk), F(IN_AI_Wv),
           F(IN_AI_Wo), F(IN_AI_Wb), F(IN_AB_W1), F(IN_AB_b1), F(IN_AB_W2),
           iScA, iIntA, iBypA};
  bw[1] = {F(IN_PS_Wt), F(IN_PS_bt), F(IN_PS_Wc), F(IN_PS_Wo), F(IN_PS_bo),
           F(IN_PI_We_v), F(IN_PI_We_i), F(IN_PI_Wq), F(IN_PI_Wk), F(IN_PI_Wv),
           F(IN_PI_Wo), F(IN_PI_Wb), F(IN_PB_W1), F(IN_PB_b1), F(IN_PB_W2),
           iScP, iIntP, iBypP};
  const float* srcV[2] = {amp_v, ph_v};
  const float* srcI[2] = {amp_i, ph_i};
  float* dstP[2] = {amp_v, ph_v};  // overwrite in place at branch end

  const int TOKELEMS_BLOCKS = (int)(((size_t)B_ * NTOK * TOK) / 256);  // 32768
  const int SELELEMS_BLOCKS = (int)(((size_t)B_ * KSEL * TOK) / 256);  // 8192

  for (int br = 0; br < 2; br++) {
    const BranchW& w = bw[br];
    patchify_k<<<TOKELEMS_BLOCKS, 256, 0, stream>>>(srcV[br], visT);
    patchify_k<<<TOKELEMS_BLOCKS, 256, 0, stream>>>(srcI[br], irT);
    coordw_k<<<(NTOK * 128) / 256, 256, 0, stream>>>(w.Wc, cwbuf);

    // score: h = gelu([v,i,|v-i|]@Wt + bt + coords@Wc + fi@Wi); s = h@Wo+bo
    launch_gemm(stream, visT, irT, 2, TOK, TOK, w.Wt, 128, 0, hbuf, 128,
                B_ * NTOK, 128, 3 * TOK, 1.f, w.bt, w.iSc, NTOK, cwbuf, 128,
                nullptr, 0.f, nullptr, 0.f, 0, 1);
    score_reduce_k<<<(B_ * NTOK) / 256, 256, 0, stream>>>(hbuf, w.Wo_s, w.bo,
                                                          score, B_ * NTOK);
    topk_k<<<B_, 1024, 0, stream>>>(score, topi);
    gather_k<<<SELELEMS_BLOCKS, 256, 0, stream>>>(visT, irT, topi, selV, selI);

    // interact (attention over selected tokens)
    launch_gemm(stream, selV, nullptr, 0, 0, TOK, w.We_v, 128, 0, qin, 128,
                B_ * KSEL, 128, TOK, 1.f, nullptr, w.iInt, KSEL);
    launch_gemm(stream, selI, nullptr, 0, 0, TOK, w.We_i, 128, 0, kin, 128,
                B_ * KSEL, 128, TOK);
    launch_gemm(stream, qin, nullptr, 0, 0, 128, w.Wq, 128, 0, qm, 128,
                B_ * KSEL, 128, 128);
    launch_gemm(stream, kin, nullptr, 0, 0, 128, w.Wk, 128, 0, km, 128,
                B_ * KSEL, 128, 128);
    launch_gemm(stream, kin, nullptr, 0, 0, 128, w.Wv, 128, 0, vm, 128,
                B_ * KSEL, 128, 128);
    const float iscale = 0.17677669529663687f;  // 1/sqrt(HD)
    for (int b = 0; b < B_; b++)
      for (int h = 0; h < HEADS_; h++) {
        const float* qp = qm + ((size_t)b * KSEL) * 128 + h * HD_;
        const float* kp = km + ((size_t)b * KSEL) * 128 + h * HD_;
        float* sp = att + ((size_t)(b * HEADS_ + h)) * KSEL * KSEL;
        launch_gemm(stream, qp, nullptr, 0, 0, 128, kp, 128, 1, sp, KSEL, KSEL,
                    KSEL, HD_, iscale);
      }
    softmax_k<<<B_ * HEADS_ * KSEL, 256, 0, stream>>>(att);
    for (int b = 0; b < B_; b++)
      for (int h = 0; h < HEADS_; h++) {
        const float* ap = att + ((size_t)(b * HEADS_ + h)) * KSEL * KSEL;
        const float* vp = vm + ((size_t)b * KSEL) * 128 + h * HD_;
        float* op = obuf + ((size_t)b * KSEL) * 128 + h * HD_;
        launch_gemm(stream, ap, nullptr, 0, 0, KSEL, vp, 128, 0, op, 128, KSEL,
                    HD_, KSEL);
      }
    launch_gemm(stream, obuf, nullptr, 0, 0, 128, w.Wo_i, 128, 0, o2, 128,
                B_ * KSEL, 128, 128);
    launch_gemm(stream, o2, nullptr, 0, 0, 128, w.Wb, TOK, 0, strong, TOK,
                B_ * KSEL, TOK, 128, 1.f, nullptr, nullptr, 1, nullptr, 128,
                selV, 1.f, nullptr, 0.f, TOK, 0);

    // bypass (all tokens) -> fusedT; then overwrite selected rows
    launch_gemm(stream, visT, irT, 1, TOK, TOK, w.W1, 128, 0, hbuf, 128,
                B_ * NTOK, 128, 2 * TOK, 1.f, w.b1, w.iByp, NTOK, nullptr, 128,
                nullptr, 0.f, nullptr, 0.f, 0, 1);
    launch_gemm(stream, hbuf, nullptr, 0, 0, 128, w.W2, TOK, 0, fusedT, TOK,
                B_ * NTOK, TOK, 128, 1.f, nullptr, nullptr, 1, nullptr, 128,
                visT, 0.5f, irT, 0.5f, TOK, 0);
    scatter_k<<<SELELEMS_BLOCKS, 256, 0, stream>>>(strong, topi, fusedT);
    unpatchify_k<<<TOKELEMS_BLOCKS, 256, 0, stream>>>(fusedT, dstP[br]);
  }

  // ---- spec -> inverse FFT2 -> real ----
  spec_k<<<(int)(PLANE_ / 256), 256, 0, stream>>>(amp_v, ph_v, t0, t1);
  float* r0 = visT;
  float* r1 = irT;
  float* r2 = fusedT;
  fft_pass<<<fgrid, 128, 0, stream>>>(t0, t1, amp_i, ph_i, 1.f);
  fft_real<<<fgrid, 128, 0, stream>>>(amp_i, ph_i, r0);

  // ---- conv -> relu -> conv ----
  dim3 cgrid(W_ / 16, H_ / 16, B_);
  conv3x3_k<<<cgrid, dim3(16, 16), 0, stream>>>(r0, F(IN_Wc1), F(IN_bc1), r1, 1);
  conv3x3_k<<<cgrid, dim3(16, 16), 0, stream>>>(r1, F(IN_Wc2), F(IN_bc2), r2, 0);

  // ---- affine epilogue + residual ----
  final_k<<<(int)(PLANE_ / 256), 256, 0, stream>>>(r2, gamma, beta, vis, ir,
                                                   (float*)d_out);
}